// EWMA_model2_40200893890671
// MI455X (gfx1250) — compile-verified
//
#include <hip/hip_runtime.h>
#include <math.h>

#define NP    256          // assets (p)
#define LAGN  24
#define STEPW 20
#define LW    (LAGN*STEPW) // 480 window length
#define TW    1000         // number of windows
#define KC    32           // K-chunk staged in LDS
#define NCH   (LW/KC)      // 15 chunks
#define PST   544          // LDS dword stride per K-pair (2*NP + 32; PST%64==32 -> conflict-free)
#define SST   257          // LDS stride for Sigma (pad vs 256)
#define CGIT  64           // CG iterations (fixed, deterministic)

typedef __attribute__((ext_vector_type(2))) float v2f;
typedef __attribute__((ext_vector_type(4))) float f4;
typedef __attribute__((ext_vector_type(8))) float v8f;

// ---------------- kernel 1: EWMA weights -> sqrt(w) + scale factor ----------------
__global__ __launch_bounds__(512) void k_weights(const float* __restrict__ a,
                                                 float* __restrict__ ws) {
  __shared__ float red[512];
  __shared__ float sw[LW];
  const int tid = threadIdx.x;
  const float av = a[0];
  float wraw = 0.0f;
  if (tid < LW) { wraw = __powf(av, (float)(LW - 1 - tid)); sw[tid] = wraw; }
  red[tid] = wraw; __syncthreads();
  for (int s = 256; s > 0; s >>= 1) { if (tid < s) red[tid] += red[tid + s]; __syncthreads(); }
  const float scale = (float)LW / red[0];
  __syncthreads();
  float wn = 0.0f;
  if (tid < LW) {
    wn = sw[tid] * scale;
    ws[tid] = sqrtf(wn);          // sqrt(w): folded symmetrically into both GEMM operands
  }
  red[tid] = wn * wn; __syncthreads();
  for (int s = 256; s > 0; s >>= 1) { if (tid < s) red[tid] += red[tid + s]; __syncthreads(); }
  if (tid == 0) {
    const float denom = 1.0f - red[0] / ((float)LW * (float)LW);
    ws[LW] = 1.0f / ((float)LW * denom);   // inv = 1/(L*denom)
  }
}

// ---------------- kernel 2: per-window Sigma (WMMA f32) + CG solve ----------------
__global__ __launch_bounds__(512) void k_main(const float* __restrict__ Y,
                                              const float* __restrict__ ws,
                                              float* __restrict__ rese,
                                              float* __restrict__ ress) {
  __shared__ __align__(16) float sSig[NP * SST];       // 263168 B
  __shared__ __align__(16) float sY2[(KC/2) * PST];    //  34816 B, K-pair interleaved
  __shared__ float sQ[LW];                             // sqrt(w)
  __shared__ float sMean[2 * NP];
  __shared__ float sVx[NP];
  __shared__ float sVr[NP];
  __shared__ float sVp[NP];
  __shared__ float sVap[NP];
  __shared__ float sRed[512];
  __shared__ float sRets[STEPW];

  const int t    = blockIdx.x;
  const int tid  = threadIdx.x;
  const int lane = tid & 31;
  const int wv   = tid >> 5;     // wave 0..15
  const int hl   = lane >> 4;    // half-wave select
  const int li   = lane & 15;
  const int wr   = wv >> 2;      // wave row in 4x4 wave grid
  const int wc   = wv & 3;       // wave col
  const long start = (long)t * STEPW;

  for (int i = tid; i < LW; i += 512) sQ[i] = ws[i];
  const float inv = ws[LW];

  v8f acc[4][4];
#pragma unroll
  for (int i = 0; i < 4; ++i)
#pragma unroll
    for (int j = 0; j < 4; ++j)
      acc[i][j] = (v8f){0.f,0.f,0.f,0.f,0.f,0.f,0.f,0.f};

  const int colm = tid & (NP - 1);
  const int seg  = tid >> 8;          // 0 or 1
  float mpart = 0.0f;

  // ---- weighted Gram accumulation: S = (sqrtW Y)^T (sqrtW Y) = sum_l w_l y_l y_l^T ----
  for (int c = 0; c < NCH; ++c) {
    __syncthreads();
    const float* gy = Y + (start + (long)c * KC) * NP;
    // stage chunk: K-pair interleaved layout, row pair scaled by sqrt(w)
    for (int u = tid; u < (KC / 2) * (NP / 4); u += 512) {
      const int kp   = u >> 6;        // k-pair 0..15
      const int q    = u & 63;        // 4-col group
      const int col0 = q * 4;
      f4 v0 = ((const f4*)(gy + (2 * kp)     * NP))[q];
      f4 v1 = ((const f4*)(gy + (2 * kp + 1) * NP))[q];
      v0 *= sQ[c * KC + 2 * kp];
      v1 *= sQ[c * KC + 2 * kp + 1];
      float* dst = sY2 + kp * PST + col0 * 2;
#pragma unroll
      for (int m = 0; m < 4; ++m) {
        v2f pr; pr.x = v0[m]; pr.y = v1[m];
        *((v2f*)(dst + m * 2)) = pr;
      }
    }
    // prefetch next chunk's global data (gfx1250 global_prefetch_b8)
    if (c + 1 < NCH) {
      const float* gn = Y + (start + (long)(c + 1) * KC) * NP;
      if (tid < 256) __builtin_prefetch((const void*)(gn + tid * 32), 0, 0);
    }
    __syncthreads();

    // weighted mean partials: sqrt(w) * (sqrt(w)*Y) = w*Y  (2 threads per column)
    {
#pragma unroll
      for (int kp = 0; kp < 8; ++kp) {
        const int kpg = seg * 8 + kp;
        const v2f pr = *((const v2f*)(sY2 + kpg * PST + colm * 2));
        mpart += sQ[c * KC + 2 * kpg] * pr.x + sQ[c * KC + 2 * kpg + 1] * pr.y;
      }
    }

    // WMMA: 8 K-steps of 4 per chunk; each wave does a 4x4 grid of 16x16 tiles
#pragma unroll
    for (int ks = 0; ks < KC / 4; ++ks) {
      const int kp = ks * 2 + hl;               // per-half-wave K pair
      const float* yp = sY2 + kp * PST;
      v2f af[4], bf[4];
#pragma unroll
      for (int i = 0; i < 4; ++i)               // A fragments: single aligned b64 each
        af[i] = *((const v2f*)(yp + (wr * 64 + i * 16 + li) * 2));
#pragma unroll
      for (int j = 0; j < 4; ++j)               // B fragments
        bf[j] = *((const v2f*)(yp + (wc * 64 + j * 16 + li) * 2));
#pragma unroll
      for (int i = 0; i < 4; ++i)
#pragma unroll
        for (int j = 0; j < 4; ++j)
          acc[i][j] = __builtin_amdgcn_wmma_f32_16x16x4_f32(
              false, af[i], false, bf[j], (short)0, acc[i][j], false, false);
    }
  }

  // ---- finalize weighted mean ----
  __syncthreads();
  sMean[seg * NP + colm] = mpart;
  __syncthreads();
  if (tid < NP) sMean[tid] = (sMean[tid] + sMean[NP + tid]) * (1.0f / (float)LW);
  __syncthreads();

  // ---- epilogue: Sigma = (S - L*m m^T) * inv  -> LDS ----
#pragma unroll
  for (int i = 0; i < 4; ++i) {
    const int rbase = wr * 64 + i * 16 + 8 * hl;
#pragma unroll
    for (int j = 0; j < 4; ++j) {
      const int col = wc * 64 + j * 16 + li;
      const float mc = sMean[col];
#pragma unroll
      for (int r = 0; r < 8; ++r) {
        const int row = rbase + r;
        sSig[row * SST + col] = (acc[i][j][r] - (float)LW * sMean[row] * mc) * inv;
      }
    }
  }
  __syncthreads();

  // ---- CG solve: Sigma x = 1 (SPD), fixed 64 iterations ----
  if (tid < NP) { sVx[tid] = 0.0f; sVr[tid] = 1.0f; sVp[tid] = 1.0f; }
  float rs = (float)NP;
  for (int it = 0; it < CGIT; ++it) {
    __syncthreads();
    // Ap = Sigma * p  (2 threads per row, 128 MACs each)
    float partial = 0.0f;
    {
      const float* srow = sSig + colm * SST + seg * 128;
      const float* pv   = sVp + seg * 128;
#pragma unroll 8
      for (int k = 0; k < 128; ++k) partial += srow[k] * pv[k];
    }
    sRed[tid] = partial;
    __syncthreads();
    if (tid < NP) sVap[tid] = sRed[tid] + sRed[tid + NP];
    __syncthreads();
    // pAp
    sRed[tid] = (tid < NP) ? sVp[tid] * sVap[tid] : 0.0f;
    __syncthreads();
    for (int s = 256; s > 0; s >>= 1) { if (tid < s) sRed[tid] += sRed[tid + s]; __syncthreads(); }
    const float alpha = rs / sRed[0];
    __syncthreads();
    float rv = 0.0f;
    if (tid < NP) {
      sVx[tid] += alpha * sVp[tid];
      rv = sVr[tid] - alpha * sVap[tid];
      sVr[tid] = rv;
    }
    sRed[tid] = rv * rv;
    __syncthreads();
    for (int s = 256; s > 0; s >>= 1) { if (tid < s) sRed[tid] += sRed[tid + s]; __syncthreads(); }
    const float rsn = sRed[0];
    const float beta = rsn / rs;
    rs = rsn;
    __syncthreads();
    if (tid < NP) sVp[tid] = sVr[tid] + beta * sVp[tid];
  }

  // ---- portfolio returns on the next STEP rows ----
  __syncthreads();
  sRed[tid] = (tid < NP) ? sVx[tid] : 0.0f;
  __syncthreads();
  for (int s = 256; s > 0; s >>= 1) { if (tid < s) sRed[tid] += sRed[tid + s]; __syncthreads(); }
  const float sx = sRed[0];
  if (tid < STEPW) {
    const float* yte = Y + (start + LW + tid) * NP;
    float d = 0.0f;
    for (int p2 = 0; p2 < NP; ++p2) d += yte[p2] * sVx[p2];
    sRets[tid] = d / sx;
  }
  __syncthreads();
  if (tid == 0) {
    float se = 0.0f;
    for (int s = 0; s < STEPW; ++s) se += sRets[s];
    const float m = se / (float)STEPW;
    float ssv = 0.0f;
    for (int s = 0; s < STEPW; ++s) { const float d2 = sRets[s] - m; ssv += d2 * d2; }
    ssv /= (float)(STEPW - 1);
    rese[t] = se;
    ress[t] = ssv;
  }
}

// ---------------- kernel 3: final reduction -> (vol, mu, sharpe) ----------------
__global__ __launch_bounds__(512) void k_final(const float* __restrict__ rese,
                                               const float* __restrict__ ress,
                                               float* __restrict__ out) {
  __shared__ float r1[512];
  __shared__ float r2[512];
  const int tid = threadIdx.x;
  float a1 = 0.0f, a2 = 0.0f;
  for (int i = tid; i < TW; i += 512) { a1 += rese[i]; a2 += ress[i]; }
  r1[tid] = a1; r2[tid] = a2; __syncthreads();
  for (int s = 256; s > 0; s >>= 1) {
    if (tid < s) { r1[tid] += r1[tid + s]; r2[tid] += r2[tid + s]; }
    __syncthreads();
  }
  if (tid == 0) {
    const float mu_e = r1[0] / (float)TW;
    const float ms   = r2[0] / (float)TW;
    const float vol  = sqrtf(ms * 252.0f);
    const float mu   = mu_e / (float)STEPW * 252.0f;
    out[0] = vol;
    out[1] = mu;
    out[2] = mu / vol;
  }
}

extern "C" void kernel_launch(void* const* d_in, const int* in_sizes, int n_in,
                              void* d_out, int out_size, void* d_ws, size_t ws_size,
                              hipStream_t stream) {
  const float* Y = (const float*)d_in[0];
  const float* a = (const float*)d_in[1];
  // d_in[2] == b is unused by the reference computation
  float* ws   = (float*)d_ws;        // [0..479] sqrt(w), [480] inv scale
  float* rese = ws + 512;            // [512 .. 512+999]
  float* ress = ws + 512 + TW;       // [1512 .. 1512+999]
  float* out  = (float*)d_out;

  k_weights<<<1, 512, 0, stream>>>(a, ws);
  k_main  <<<TW, 512, 0, stream>>>(Y, ws, rese, ress);
  k_final <<<1, 512, 0, stream>>>(rese, ress, out);
}